// SAMSLoss_58334245814721
// MI455X (gfx1250) — compile-verified
//
#include <hip/hip_runtime.h>
#include <hip/hip_bf16.h>
#include <math.h>

#define HW   256
#define NPIX 65536
#define BIGF 1.0e4f

typedef __attribute__((ext_vector_type(2))) float v2f;
typedef __attribute__((ext_vector_type(8))) float v8f;

// ---------------------------------------------------------------------------
// Pass 1 EDT (columns): f[i,j] = min over i' with G[i',j]>0.5 of |i-i'| (BIG if none)
// One block per (image, column). Column staged in LDS; writes f^2.
// ---------------------------------------------------------------------------
__global__ void k_col_edt(const float* __restrict__ target, float* __restrict__ f2buf) {
  __shared__ float col[HW];
  const int b = blockIdx.x >> 8;
  const int j = blockIdx.x & 255;
  const int i = threadIdx.x;
  const float* G = target + ((size_t)b * 2 + 1) * NPIX;
  // col[ip] = 0 if pixel in set, BIG otherwise; then cand = max(|i-ip|, col[ip])
  col[i] = (G[i * HW + j] > 0.5f) ? 0.0f : BIGF;
  __syncthreads();
  const float fi = (float)i;
  float m = BIGF;
#pragma unroll 8
  for (int ip = 0; ip < HW; ++ip) {
    float d = fabsf(fi - (float)ip);        // <= 255 < BIG
    m = fminf(m, fmaxf(d, col[ip]));        // mask: d ; no-mask: BIG
  }
  f2buf[(size_t)b * NPIX + (size_t)i * HW + j] = m * m;
}

// ---------------------------------------------------------------------------
// Pass 2 EDT (rows): d1[i,j] = sqrt(min over j' of f2[i,j'] + (j-j')^2), in place.
// Row staged into LDS via CDNA5 async global->LDS load.
// ---------------------------------------------------------------------------
__global__ void k_row_edt(float* __restrict__ f2buf) {
  __shared__ float f2[HW];
  const int bi = blockIdx.x;                 // b*256 + i
  const int j  = threadIdx.x;
  float* row = f2buf + (size_t)bi * HW;
  // async copy: 4 bytes from the global address row + j*4 into LDS at f2 + j*4
  unsigned lds  = (unsigned)(uintptr_t)(&f2[0]) + (unsigned)(j * 4);
  unsigned voff = (unsigned)(j * 4);
  unsigned long long sbase = (unsigned long long)(uintptr_t)row;
  asm volatile("global_load_async_to_lds_b32 %0, %1, %2"
               :: "v"(lds), "v"(voff), "s"(sbase) : "memory");
  asm volatile("s_wait_asynccnt 0" ::: "memory");
  __syncthreads();
  const float jf = (float)j;
  float m = 3.4e38f;
#pragma unroll 8
  for (int jp = 0; jp < HW; ++jp) {
    float d = jf - (float)jp;
    m = fminf(m, fmaf(d, d, f2[jp]));
  }
  row[j] = sqrtf(m);                         // overwrite with d1 (safe: row fully in LDS)
}

// ---------------------------------------------------------------------------
// Per-image stats: count(G==1), count(G==0), count(G>0.5), min/max of H0raw.
// Also writes H0raw (clamped haematoxylin channel) to h0buf.
// ---------------------------------------------------------------------------
__global__ void k_stats(const float* __restrict__ target, const float* __restrict__ img,
                        float* __restrict__ h0buf, float* __restrict__ stats,
                        float c0, float c1, float c2, float invLA) {
  const int b = blockIdx.x;
  const int t = threadIdx.x;
  const float* G = target + ((size_t)b * 2 + 1) * NPIX;
  const float* R = img + (size_t)b * 3 * NPIX;
  float* H = h0buf + (size_t)b * NPIX;
  float cnt1 = 0.f, cnt0 = 0.f, cntm = 0.f;
  float hmin = 3.4e38f, hmax = -3.4e38f;
  for (int p = t; p < NPIX; p += 256) {
    float g = G[p];
    cnt1 += (g == 1.0f) ? 1.f : 0.f;
    cnt0 += (g == 0.0f) ? 1.f : 0.f;
    cntm += (g > 0.5f)  ? 1.f : 0.f;
    float r  = fmaxf(R[p],            1e-6f);
    float gg = fmaxf(R[p + NPIX],     1e-6f);
    float bb = fmaxf(R[p + 2 * NPIX], 1e-6f);
    float h = logf(r) * invLA * c0 + logf(gg) * invLA * c1 + logf(bb) * invLA * c2;
    h = fmaxf(h, 0.0f);
    H[p] = h;
    hmin = fminf(hmin, h);
    hmax = fmaxf(hmax, h);
  }
  __shared__ float s1[256], s0[256], sm[256], smin[256], smax[256];
  s1[t] = cnt1; s0[t] = cnt0; sm[t] = cntm; smin[t] = hmin; smax[t] = hmax;
  __syncthreads();
  for (int off = 128; off > 0; off >>= 1) {
    if (t < off) {
      s1[t] += s1[t + off]; s0[t] += s0[t + off]; sm[t] += sm[t + off];
      smin[t] = fminf(smin[t], smin[t + off]);
      smax[t] = fmaxf(smax[t], smax[t + off]);
    }
    __syncthreads();
  }
  if (t == 0) {
    float* S = stats + b * 8;
    S[0] = s1[0]; S[1] = s0[0]; S[2] = sm[0]; S[3] = smin[0]; S[4] = smax[0];
  }
}

// ---------------------------------------------------------------------------
// Row sums of preds[:,1] via WMMA: rowsum = inp @ ones(256).
// One wave (32 lanes) per 16-row group; A = 16x4 f32 tiles, B = 4x16 ones.
// D[:,0] after 64 accumulating WMMAs = row sums.
// ---------------------------------------------------------------------------
__global__ void k_rowsum(const float* __restrict__ preds, float* __restrict__ rowsum) {
  const int b = blockIdx.x >> 4;
  const int g = blockIdx.x & 15;
  const float* base = preds + ((size_t)b * 2 + 1) * NPIX + (size_t)g * 16 * HW;
  const int lane = threadIdx.x;              // 0..31
  const int M  = lane & 15;
  const int kh = (lane >> 4) << 1;           // 0 (lanes 0-15) or 2 (lanes 16-31)
  v8f c = {};
  v2f bm; bm.x = 1.0f; bm.y = 1.0f;          // all-ones B, layout-invariant
  for (int k0 = 0; k0 < HW; k0 += 4) {
    const float* p = base + M * HW + k0 + kh;
    v2f a; a.x = p[0]; a.y = p[1];
    c = __builtin_amdgcn_wmma_f32_16x16x4_f32(false, a, false, bm,
                                              (short)0, c, false, false);
  }
  // N=0 column: lane 0 holds M=0..7 in c[0..7]; lane 16 holds M=8..15
  if ((lane & 15) == 0) {
    float* out = rowsum + b * HW + g * 16 + (lane >> 4) * 8;
#pragma unroll
    for (int r = 0; r < 8; ++r) out[r] = c[r];
  }
}

// ---------------------------------------------------------------------------
// Fused per-pixel loss + block tree-reduce (deterministic; no float atomics).
// ---------------------------------------------------------------------------
__global__ void k_loss(const float* __restrict__ preds, const float* __restrict__ target,
                       const float* __restrict__ d1buf, const float* __restrict__ h0buf,
                       const float* __restrict__ rowsum, const float* __restrict__ stats,
                       float* __restrict__ partials) {
  const int bi = blockIdx.x;                 // b*256 + i
  const int b  = bi >> 8;
  const int i  = bi & 255;
  const int j  = threadIdx.x;
  const float* S = stats + b * 8;
  const float ones = S[0], zeros = S[1], cntm = S[2], hmin = S[3], hmax = S[4];
  const float sc1 = (ones  > 0.f) ? (1.f / fmaxf(ones,  1.f)) : 1.f;
  const float sc0 = (zeros > 0.f) ? (1.f / fmaxf(zeros, 1.f)) : 1.f;
  // masked==0 set is full grid iff every pixel is in the mask, else empty -> d2 const
  const float d2 = (cntm >= 65536.0f) ? 0.0f : BIGF;

  const size_t pix = (size_t)bi * HW + j;    // = b*NPIX + i*HW + j
  const float G   = target[((size_t)b * 2 + 1) * NPIX + (size_t)i * HW + j];
  const float h0  = h0buf[pix];
  const float hn  = (hmax > hmin) ? (h0 - hmin) / (hmax - hmin) : h0;
  const float H1  = (1.0f - hn) * G;
  const float H2  = hn * (1.0f - G);
  const float wcv = (G == 1.0f) ? sc1 : -sc0;
  const float wz  = wcv + H1 * H1 * H1 + H2 * sqrtf(H2);   // H2^1.5, H2 >= 0
  const float d1  = d1buf[pix];
  const float s   = d1 + d2;
  const float w   = wz + 10.0f * expf(-(s * s) * (1.0f / 32.0f));

  const float inp = preds[((size_t)b * 2 + 1) * NPIX + (size_t)i * HW + j];
  float pr = inp / rowsum[bi];
  pr = fminf(fmaxf(pr, 1e-7f), 1.0f - 1e-7f);
  float contrib = -w * logf(pr);

  __shared__ float red[256];
  red[j] = contrib;
  __syncthreads();
  for (int off = 128; off > 0; off >>= 1) {
    if (j < off) red[j] += red[j + off];
    __syncthreads();
  }
  if (j == 0) partials[bi] = red[0];
}

__global__ void k_final(const float* __restrict__ partials, float* __restrict__ out, int n) {
  const int t = threadIdx.x;
  float acc = 0.f;
  for (int p = t; p < n; p += 256) acc += partials[p];
  __shared__ float red[256];
  red[t] = acc;
  __syncthreads();
  for (int off = 128; off > 0; off >>= 1) {
    if (t < off) red[t] += red[t + off];
    __syncthreads();
  }
  if (t == 0) out[0] = red[0];
}

// ---------------------------------------------------------------------------
extern "C" void kernel_launch(void* const* d_in, const int* in_sizes, int n_in,
                              void* d_out, int out_size, void* d_ws, size_t ws_size,
                              hipStream_t stream) {
  const float* preds  = (const float*)d_in[0];
  const float* target = (const float*)d_in[1];
  const float* img    = (const float*)d_in[2];
  float* out = (float*)d_out;
  const int B = in_sizes[0] / (2 * NPIX);    // 16

  float* ws       = (float*)d_ws;
  float* f2buf    = ws;                          // B*NPIX; holds f^2 then d1
  float* h0buf    = f2buf + (size_t)B * NPIX;    // B*NPIX
  float* rowsum   = h0buf + (size_t)B * NPIX;    // B*256
  float* stats    = rowsum + (size_t)B * HW;     // B*8
  float* partials = stats + (size_t)B * 8;       // B*256

  // Column 0 of inv(rgb_from_hed) (entries rounded to f32 first, as in numpy)
  const double m00 = (double)0.65f, m01 = (double)0.70f, m02 = (double)0.29f;
  const double m10 = (double)0.07f, m11 = (double)0.99f, m12 = (double)0.11f;
  const double m20 = (double)0.27f, m21 = (double)0.57f, m22 = (double)0.78f;
  const double det = m00 * (m11 * m22 - m12 * m21)
                   - m01 * (m10 * m22 - m12 * m20)
                   + m02 * (m10 * m21 - m11 * m20);
  const float c0 = (float)( (m11 * m22 - m12 * m21) / det);
  const float c1 = (float)(-(m10 * m22 - m12 * m20) / det);
  const float c2 = (float)( (m10 * m21 - m11 * m20) / det);
  const float invLA = (float)(1.0 / log(1e-6));  // 1/LOG_ADJUST

  k_col_edt<<<B * HW, HW, 0, stream>>>(target, f2buf);
  k_stats  <<<B,      HW, 0, stream>>>(target, img, h0buf, stats, c0, c1, c2, invLA);
  k_row_edt<<<B * HW, HW, 0, stream>>>(f2buf);
  k_rowsum <<<B * 16, 32, 0, stream>>>(preds, rowsum);
  k_loss   <<<B * HW, HW, 0, stream>>>(preds, target, f2buf, h0buf, rowsum, stats, partials);
  k_final  <<<1,      HW, 0, stream>>>(partials, out, B * HW);
}